// MultiscaleResidualVectorQuantize_18992345382932
// MI455X (gfx1250) — compile-verified
//
#include <hip/hip_runtime.h>
#include <math.h>

#define B_   8
#define D_   512
#define T_   8192
#define NQ   16
#define CBS  1024
#define CBD  8
#define TILE 16
#define NTILES 4096            // B_*T_/TILE
#define ZQ_ELEMS 33554432ULL   // B_*D_*T_
#define LOSS_DEN 524288.0f     // B_*8*T_

typedef float v2f __attribute__((ext_vector_type(2)));
typedef float v8f __attribute__((ext_vector_type(8)));

// ---------------- prep: weight-norm in_proj weights, padded to 16 rows ----
__global__ __launch_bounds__(256) void k_prep_inw(const float* __restrict__ in_v,
                                                  const float* __restrict__ in_g,
                                                  float* __restrict__ in_w) {
  int q = blockIdx.x;                 // stage
  int tid = threadIdx.x;
  int r = tid >> 5, lane = tid & 31;  // 8 waves, one row each
  const float* vrow = in_v + (size_t)(q * 8 + r) * 512;
  float ss = 0.f;
  for (int c = lane; c < 512; c += 32) { float v = vrow[c]; ss += v * v; }
#pragma unroll
  for (int o = 16; o >= 1; o >>= 1) ss += __shfl_xor(ss, o, 32);
  float scale = in_g[q * 8 + r] / sqrtf(ss);
  float* wrow = in_w + (size_t)q * 8192 + (size_t)r * 512;
  for (int c = lane; c < 512; c += 32) wrow[c] = vrow[c] * scale;
  // zero-pad rows 8..15 so WMMA A-operand (M padded to 16) needs no masking
  float* wpad = in_w + (size_t)q * 8192 + 4096;
  for (int j = tid; j < 4096; j += 256) wpad[j] = 0.f;
}

// ------- prep: out_proj weight-norm rows, l2-normalized codebooks, zero partials
__global__ __launch_bounds__(256) void k_prep_rows(const float* __restrict__ out_v,
                                                   const float* __restrict__ out_g,
                                                   const float* __restrict__ cb,
                                                   float* __restrict__ out_w,
                                                   float* __restrict__ cbn,
                                                   float* __restrict__ partial) {
  int idx = blockIdx.x * 256 + threadIdx.x;
  if (idx < NTILES) partial[idx] = 0.f;
  if (idx < NQ * 512) {
    const float* vr = out_v + (size_t)idx * 8;
    float ss = 0.f;
#pragma unroll
    for (int c = 0; c < 8; ++c) ss += vr[c] * vr[c];
    float scale = out_g[idx] / sqrtf(ss);
    float* wr = out_w + (size_t)idx * 8;
#pragma unroll
    for (int c = 0; c < 8; ++c) wr[c] = vr[c] * scale;
  } else if (idx < NQ * 512 + NQ * CBS) {
    int j = idx - NQ * 512;
    const float* vr = cb + (size_t)j * 8;
    float ss = 0.f;
#pragma unroll
    for (int c = 0; c < 8; ++c) ss += vr[c] * vr[c];
    float n = sqrtf(ss); if (n < 1e-12f) n = 1e-12f;
    float inv = 1.f / n;
    float* wr = cbn + (size_t)j * 8;
#pragma unroll
    for (int c = 0; c < 8; ++c) wr[c] = vr[c] * inv;
  }
}

// ---------------- init: residual = z (stored in d_out), z_e stage0 ----------
__global__ __launch_bounds__(128) void k_init(const float* __restrict__ z,
                                              const float* __restrict__ in_w0,
                                              const float* __restrict__ in_b0,
                                              float* __restrict__ residual,
                                              float* __restrict__ z_e) {
  __shared__ float lds[512 * 16];
  int tid = threadIdx.x;
  int tile = blockIdx.x;
  int b = (tile * TILE) / T_;
  int t0 = (tile * TILE) % T_;
  for (int j = tid; j < 8192; j += 128) {
    int c = j >> 4, col = j & 15;
    size_t g = (size_t)(b * 512 + c) * T_ + t0 + col;
    float v = z[g];
    lds[j] = v;
    residual[g] = v;
  }
  __syncthreads();
  int col = tid >> 3, m = tid & 7;
  const float* wr = in_w0 + (size_t)m * 512;
  float acc = in_b0[m];
  for (int c = 0; c < 512; ++c) acc += wr[c] * lds[c * 16 + col];
  z_e[(size_t)(b * 8 + m) * T_ + t0 + col] = acc;
}

// ---------------- adaptive average pool T -> S ------------------------------
__global__ __launch_bounds__(256) void k_pool(const float* __restrict__ z_e,
                                              float* __restrict__ inter, int S) {
  int idx = blockIdx.x * 256 + threadIdx.x;
  if (idx >= 8 * 8 * S) return;
  int s = idx % S;
  int bc = idx / S;
  int start = (s * T_) / S;
  int end = ((s + 1) * T_ + S - 1) / S;
  const float* row = z_e + (size_t)bc * T_;
  float acc = 0.f;
  for (int t = start; t < end; ++t) acc += row[t];
  inter[(size_t)bc * S + s] = acc / (float)(end - start);
}

// ------- codebook lookup (argmax of dot with l2-normed cb) + gather ---------
__global__ __launch_bounds__(256) void k_lookup(const float* __restrict__ inter,
                                                const float* __restrict__ cbn_s,
                                                const float* __restrict__ cb_s,
                                                float* __restrict__ q_s,
                                                int S, int nSb) {
  __shared__ float lcb[CBS * 8];
  int b = blockIdx.x / nSb;
  int chunk = blockIdx.x % nSb;
  int tid = threadIdx.x;
  for (int j = tid; j < CBS * 8; j += 256) lcb[j] = cbn_s[j];
  __syncthreads();
  int s = chunk * 256 + tid;
  if (s >= S) return;
  float v[8];
#pragma unroll
  for (int c = 0; c < 8; ++c) v[c] = inter[(size_t)(b * 8 + c) * S + s];
  float best = -3.4e38f; int bi = 0;
  for (int j = 0; j < CBS; ++j) {
    const float* e = &lcb[j * 8];
    float d = v[0]*e[0] + v[1]*e[1] + v[2]*e[2] + v[3]*e[3]
            + v[4]*e[4] + v[5]*e[5] + v[6]*e[6] + v[7]*e[7];
    if (d > best) { best = d; bi = j; }   // strict > : first max == argmin tie rule
  }
  const float* q = cb_s + (size_t)bi * 8;
#pragma unroll
  for (int c = 0; c < 8; ++c) q_s[(size_t)(b * 8 + c) * S + s] = q[c];
}

// ---- fused stage: upsample + Phi + loss + out_proj(WMMA) + residual update
// ---- + next-stage in_proj (WMMA, K=512) ------------------------------------
__global__ __launch_bounds__(128) void k_stage(const float* __restrict__ q_s,
                                               const float* __restrict__ z_e_cur,
                                               float* __restrict__ z_e_next,
                                               float* __restrict__ residual,
                                               const float* __restrict__ out_w_s,
                                               const float* __restrict__ out_b_s,
                                               const float* __restrict__ in_w_next,
                                               const float* __restrict__ in_b_next,
                                               const float* __restrict__ phi_w_p,
                                               const float* __restrict__ phi_b_p,
                                               float* __restrict__ partial,
                                               int S, int last) {
  __shared__ float qup[24 * 9];     // upsampled halo positions, padded stride
  __shared__ float qphi[16 * 9];    // post-Phi tile, padded stride
  __shared__ float phiw[8 * 8 * 9];
  __shared__ float rlds[512 * 17];  // updated residual tile (padded)
  __shared__ float red[4 * 8 * 32]; // cross-wave WMMA reduction
  __shared__ float lred[128];

  int tid = threadIdx.x;
  int tile = blockIdx.x;
  int b = (tile * TILE) / T_;
  int t0 = (tile * TILE) % T_;

  // prefetch the 512x16 residual tile (one 64B cacheline per row) so the
  // phase-3 global loads hit warm cache; overlaps with phases 1-2 compute.
  // Lowers to global_prefetch_b8 on gfx1250.
  for (int j = tid; j < 512; j += 128)
    __builtin_prefetch(&residual[(size_t)(b * 512 + j) * T_ + t0], 0, 3);
  // prefetch the z_e rows consumed by the loss
  if (tid < 8)
    __builtin_prefetch(&z_e_cur[(size_t)(b * 8 + tid) * T_ + t0], 0, 3);

  for (int j = tid; j < 576; j += 128) phiw[j] = phi_w_p[j];

  // phase 1: linear upsample at 24 positions (16 cols + conv halo of 4 each side)
  for (int idx = tid; idx < 24 * 8; idx += 128) {
    int pidx = idx >> 3, c = idx & 7;
    int tpos = t0 + pidx - 4;
    float val = 0.f;                       // zero padding for Phi conv
    if (tpos >= 0 && tpos < T_) {
      float pos = ((float)tpos + 0.5f) * (float)S / (float)T_ - 0.5f;
      if (pos < 0.f) pos = 0.f;
      int i0 = (int)pos; if (i0 > S - 1) i0 = S - 1;
      int i1 = (i0 + 1 < S) ? i0 + 1 : S - 1;
      float w = pos - (float)i0;
      const float* qrow = q_s + (size_t)(b * 8 + c) * S;
      val = qrow[i0] * (1.f - w) + qrow[i1] * w;
    }
    qup[pidx * 9 + c] = val;
  }
  __syncthreads();

  // phase 2: Phi conv (0.5h + 0.5(conv+b)) and squared-error accumulation
  {
    int col = tid >> 3, oc = tid & 7;
    float y = phi_b_p[oc];
#pragma unroll
    for (int k = 0; k < 9; ++k) {
      const float* qrow = &qup[(col + k) * 9];
      const float* wrow = &phiw[oc * 72 + k];
#pragma unroll
      for (int c = 0; c < 8; ++c) y += qrow[c] * wrow[c * 9];
    }
    float qv = 0.5f * qup[(col + 4) * 9 + oc] + 0.5f * y;
    qphi[col * 9 + oc] = qv;
    float d = z_e_cur[(size_t)(b * 8 + oc) * T_ + t0 + col] - qv;
    lred[tid] = d * d;
  }
  __syncthreads();
  for (int s2 = 64; s2 > 0; s2 >>= 1) {
    if (tid < s2) lred[tid] += lred[tid + s2];
    __syncthreads();
  }
  if (tid == 0) partial[blockIdx.x] += lred[0];

  // phase 3: out_proj via V_WMMA_F32_16X16X4_F32 (M=512, K=8, N=16), residual -=
  int wv = tid >> 5;          // wave 0..3
  int l = tid & 31;           // lane
  int n = l & 15;             // N (column) / M-within-tile for A
  int khalf = l >> 4;         // 0/1
  {
    int k0 = khalf * 2;
    v2f b0, b1;
    b0.x = qphi[n * 9 + k0];     b0.y = qphi[n * 9 + k0 + 1];
    b1.x = qphi[n * 9 + k0 + 4]; b1.y = qphi[n * 9 + k0 + 5];
    for (int mt = 0; mt < 8; ++mt) {
      int obase = (wv * 8 + mt) * 16;
      const float* wr = out_w_s + (size_t)(obase + n) * 8;
      v2f a0, a1;
      a0.x = wr[k0];     a0.y = wr[k0 + 1];
      a1.x = wr[k0 + 4]; a1.y = wr[k0 + 5];
      v8f acc = {};
      acc = __builtin_amdgcn_wmma_f32_16x16x4_f32(false, a0, false, b0, (short)0, acc, false, false);
      acc = __builtin_amdgcn_wmma_f32_16x16x4_f32(false, a1, false, b1, (short)0, acc, false, false);
#pragma unroll
      for (int r = 0; r < 8; ++r) {
        int o = obase + r + 8 * khalf;       // C/D layout: M = r + 8*(lane/16)
        size_t g = (size_t)(b * 512 + o) * T_ + t0 + n;
        float rv = residual[g] - (acc[r] + out_b_s[o]);
        residual[g] = rv;
        rlds[o * 17 + n] = rv;
      }
    }
  }
  __syncthreads();

  // phase 4: next-stage in_proj via WMMA (M=8 padded to 16, K=512, N=16)
  if (!last) {
    int kc = wv * 128;                       // per-wave K chunk
    const float* arow = in_w_next + (size_t)n * 512;  // rows 8..15 are zero-padded
    v8f acc = {};
    for (int ks = 0; ks < 32; ++ks) {
      int Kg = kc + ks * 4 + khalf * 2;
      v2f a, bb;
      a.x = arow[Kg];  a.y = arow[Kg + 1];
      bb.x = rlds[Kg * 17 + n];  bb.y = rlds[(Kg + 1) * 17 + n];
      acc = __builtin_amdgcn_wmma_f32_16x16x4_f32(false, a, false, bb, (short)0, acc, false, false);
    }
#pragma unroll
    for (int r = 0; r < 8; ++r) red[(wv * 8 + r) * 32 + l] = acc[r];
    __syncthreads();
    {
      int col = tid >> 3, m = tid & 7;       // element (M=m, N=col) lives at vgpr m, lane col
      float v = in_b_next[m];
#pragma unroll
      for (int w2 = 0; w2 < 4; ++w2) v += red[(w2 * 8 + m) * 32 + col];
      z_e_next[(size_t)(b * 8 + m) * T_ + t0 + col] = v;
    }
  }
}

// ---------------- finalize: z_q = z - residual (in place in d_out) ----------
__global__ __launch_bounds__(256) void k_final_zq(const float* __restrict__ z,
                                                  float* __restrict__ out) {
  size_t idx = (size_t)blockIdx.x * 256 + threadIdx.x;
  if (idx < ZQ_ELEMS) out[idx] = z[idx] - out[idx];
}

__global__ __launch_bounds__(256) void k_final_loss(const float* __restrict__ partial,
                                                    float* __restrict__ lo) {
  __shared__ float red[256];
  int tid = threadIdx.x;
  float a = 0.f;
  for (int j = 0; j < NTILES / 256; ++j) a += partial[tid + j * 256];
  red[tid] = a;
  __syncthreads();
  for (int s = 128; s > 0; s >>= 1) { if (tid < s) red[tid] += red[tid + s]; __syncthreads(); }
  if (tid == 0) { float c = red[0] / LOSS_DEN; lo[0] = c; lo[1] = c; }
}

extern "C" void kernel_launch(void* const* d_in, const int* in_sizes, int n_in,
                              void* d_out, int out_size, void* d_ws, size_t ws_size,
                              hipStream_t stream) {
  (void)in_sizes; (void)n_in; (void)out_size;
  const float* z     = (const float*)d_in[0];
  const float* in_v  = (const float*)d_in[1];
  const float* in_g  = (const float*)d_in[2];
  const float* in_b  = (const float*)d_in[3];
  const float* out_v = (const float*)d_in[4];
  const float* out_g = (const float*)d_in[5];
  const float* out_b = (const float*)d_in[6];
  const float* cbs   = (const float*)d_in[7];
  const float* phi_w = (const float*)d_in[8];
  const float* phi_b = (const float*)d_in[9];

  float* ws = (float*)d_ws;
  float* in_w    = ws;                  // 16*16*512 = 131072 (padded rows)
  float* out_w   = in_w + 131072;       // 16*512*8  = 65536
  float* cbn     = out_w + 65536;       // 16*1024*8 = 131072
  float* z_e_a   = cbn + 131072;        // 524288
  float* z_e_b   = z_e_a + 524288;      // 524288
  float* inter   = z_e_b + 524288;      // 524288 (max S)
  float* q_s     = inter + 524288;      // 524288 (max S)
  float* partial = q_s + 524288;        // 4096
  if (ws_size < (size_t)(2428928) * sizeof(float)) return;

  float* residual = (float*)d_out;                 // reused as residual buffer
  float* loss_out = (float*)d_out + ZQ_ELEMS;

  static const int S_tab[NQ] = {81, 245, 409, 655, 983, 1310, 1720, 2211,
                                2703, 3358, 4014, 4669, 5488, 6307, 7208, 8192};
  static const int P_tab[NQ] = {0, 0, 0, 0, 1, 1, 1, 2, 2, 2, 3, 3, 3, 4, 4, 4};

  k_prep_inw<<<NQ, 256, 0, stream>>>(in_v, in_g, in_w);
  k_prep_rows<<<(NQ * 512 + NQ * CBS + 255) / 256, 256, 0, stream>>>(
      out_v, out_g, cbs, out_w, cbn, partial);
  k_init<<<NTILES, 128, 0, stream>>>(z, in_w, in_b, residual, z_e_a);

  float* zcur = z_e_a;
  float* znext = z_e_b;
  for (int i = 0; i < NQ; ++i) {
    int S = S_tab[i];
    int p = P_tab[i];
    int nSb = (S + 255) / 256;
    int last = (i == NQ - 1);
    k_pool<<<(8 * 8 * S + 255) / 256, 256, 0, stream>>>(zcur, inter, S);
    k_lookup<<<B_ * nSb, 256, 0, stream>>>(inter, cbn + i * CBS * 8,
                                           cbs + i * CBS * 8, q_s, S, nSb);
    const float* iwn = last ? in_w : (in_w + (size_t)(i + 1) * 8192);
    const float* ibn = last ? in_b : (in_b + (size_t)(i + 1) * 8);
    k_stage<<<NTILES, 128, 0, stream>>>(q_s, zcur, znext, residual,
                                        out_w + (size_t)i * 4096, out_b + (size_t)i * 512,
                                        iwn, ibn,
                                        phi_w + (size_t)p * 576, phi_b + (size_t)p * 8,
                                        partial, S, last);
    float* tmp = zcur; zcur = znext; znext = tmp;
  }
  k_final_zq<<<(int)((ZQ_ELEMS + 255) / 256), 256, 0, stream>>>(z, residual);
  k_final_loss<<<1, 256, 0, stream>>>(partial, loss_out);
}